// FlashAttention_14851996909855
// MI455X (gfx1250) — compile-verified
//
#include <hip/hip_runtime.h>

// FlashAttention over heads: B=4, S=4096, H=16, D=64.
// Per (b,s): S16x16 = Q(16x64) K(16x64)^T / 8 ; P = softmax_row(S); O = P V(16x64); O *= mask.
// One wave32 per (b,s) position; all matrix math on V_WMMA_F32_16X16X4_F32.
// Pure stream (no L2 reuse) -> non-temporal loads/stores; wave-private LDS ->
// wave-local s_wait_dscnt instead of a block barrier.

typedef float v2f __attribute__((ext_vector_type(2)));
typedef float v8f __attribute__((ext_vector_type(8)));

#define WAVES_PER_BLOCK 8
#define LDS_STRIDE 18            // padded row stride in dwords (even -> 8B-aligned b64 reads)

__global__ __launch_bounds__(256) void FlashAttention_head16_kernel(
    const float* __restrict__ qkv,          // [B*S][3][16][64]
    const unsigned char* __restrict__ mask, // [B*S] bool (1 byte)
    float* __restrict__ out)                // [B*S][16][64]
{
    __shared__ float lds_p[WAVES_PER_BLOCK][16 * LDS_STRIDE];

    const int lane = threadIdx.x & 31;
    const int wave = threadIdx.x >> 5;
    const int idx  = blockIdx.x * WAVES_PER_BLOCK + wave;   // flattened b*S + s

    const int row  = lane & 15;        // matrix row owned by this lane (A operand)
    const int hi   = lane >> 4;        // lane group: 0 -> K offsets {0,1}, 1 -> {2,3}
    const int koff = hi * 2;

    const float* qrow = qkv + (size_t)idx * (3 * 16 * 64) + row * 64;   // Q[row][*]
    const float* krow = qrow + 16 * 64;                                  // K[row][*]
    const float* vbas = qkv + (size_t)idx * (3 * 16 * 64) + 2 * 16 * 64; // V[16][64]

    // ---------------- S = Q * K^T  (16x16x64 via 16 chained f32 WMMAs) -------------
    // A lane layout: A[M=row][K=4kb+koff+v]  -> contiguous float2 in Q row.
    // B lane layout: B[K=4kb+koff+v][N=row] = K[row][4kb+koff+v] -> contiguous float2 in K row.
    v8f acc = {0.f, 0.f, 0.f, 0.f, 0.f, 0.f, 0.f, 0.f};
#pragma unroll
    for (int kb = 0; kb < 16; ++kb) {
        v2f a = __builtin_nontemporal_load((const v2f*)(qrow + kb * 4 + koff));
        v2f b = __builtin_nontemporal_load((const v2f*)(krow + kb * 4 + koff));
        acc = __builtin_amdgcn_wmma_f32_16x16x4_f32(false, a, false, b,
                                                    (short)0, acc, false, false);
    }

    // ---------------- row softmax (over N) inside each 16-lane half ----------------
    // C layout: VGPR r, lanes 0-15: S[r][lane]; lanes 16-31: S[r+8][lane-16].
    const float scale = 0.125f;        // 1/sqrt(64)
    float p[8];
#pragma unroll
    for (int r = 0; r < 8; ++r) {
        float s = acc[r] * scale;
        float m = s;
        m = fmaxf(m, __shfl_xor(m, 1));
        m = fmaxf(m, __shfl_xor(m, 2));
        m = fmaxf(m, __shfl_xor(m, 4));
        m = fmaxf(m, __shfl_xor(m, 8));
        float e = __expf(s - m);
        float t = e;
        t += __shfl_xor(t, 1);
        t += __shfl_xor(t, 2);
        t += __shfl_xor(t, 4);
        t += __shfl_xor(t, 8);
        p[r] = e / t;
    }

    // ---------------- C-layout -> A-layout transpose through wave-private LDS ------
    float* lp = lds_p[wave];
#pragma unroll
    for (int r = 0; r < 8; ++r) {
        const int m = r + hi * 8;
        lp[m * LDS_STRIDE + row] = p[r];
    }
    // Producer == consumer wave; LDS ops are in-order per wave. Wave-local wait is
    // sufficient -- no block-wide s_barrier needed.
    asm volatile("s_wait_dscnt 0x0" ::: "memory");

    v2f pa[4];
#pragma unroll
    for (int kb = 0; kb < 4; ++kb) {
        // A[M=row][K=4kb+koff+v] -> 8B-aligned ds_load_b64
        pa[kb] = *(const v2f*)(lp + row * LDS_STRIDE + kb * 4 + koff);
    }

    // ---------------- O = P * V (4 N-tiles x 16x16x16), mask, store ----------------
    const float mval = mask[idx] ? 1.0f : 0.0f;
    float* obase = out + (size_t)idx * (16 * 64);

#pragma unroll
    for (int nt = 0; nt < 4; ++nt) {
        const int n0 = nt * 16;
        v8f o = {0.f, 0.f, 0.f, 0.f, 0.f, 0.f, 0.f, 0.f};
#pragma unroll
        for (int kb = 0; kb < 4; ++kb) {
            const int k0 = kb * 4 + koff;
            v2f bv;
            bv.x = __builtin_nontemporal_load(vbas + (k0 + 0) * 64 + n0 + row); // B[K=k0  ][N]
            bv.y = __builtin_nontemporal_load(vbas + (k0 + 1) * 64 + n0 + row); // B[K=k0+1][N]
            o = __builtin_amdgcn_wmma_f32_16x16x4_f32(false, pa[kb], false, bv,
                                                      (short)0, o, false, false);
        }
#pragma unroll
        for (int r = 0; r < 8; ++r) {
            const int m = r + hi * 8;                // output row
            __builtin_nontemporal_store(o[r] * mval, obase + m * 64 + n0 + row);
        }
    }
}

extern "C" void kernel_launch(void* const* d_in, const int* in_sizes, int n_in,
                              void* d_out, int out_size, void* d_ws, size_t ws_size,
                              hipStream_t stream) {
    const float* qkv          = (const float*)d_in[0];
    const unsigned char* mask = (const unsigned char*)d_in[1]; // bool array, 1 byte/elem
    float* out                = (float*)d_out;

    const int total = 4 * 4096;                    // B * S independent tiles
    dim3 grid(total / WAVES_PER_BLOCK);            // 2048 blocks
    dim3 block(WAVES_PER_BLOCK * 32);              // 8 wave32 per block

    FlashAttention_head16_kernel<<<grid, block, 0, stream>>>(qkv, mask, out);

    (void)in_sizes; (void)n_in; (void)out_size; (void)d_ws; (void)ws_size;
}